// CustomTripletMarginLoss_55774445306141
// MI455X (gfx1250) — compile-verified
//
#include <hip/hip_runtime.h>
#include <hip/hip_bf16.h>
#include <math.h>

// ---------------- problem constants (from reference) ----------------
#define NROWS 8192
#define DIMS  512
#define MARGIN 0.1f
#define MINE_T2 25.0f          // (MARGIN*100/2)^2 = 5^2, compare in squared domain

#define JSPLIT 4
#define JCHUNK (NROWS / JSPLIT)          // 2048 columns per split
#define TILES_PER_SPLIT (JCHUNK / 16)    // 128 WMMA tiles per split
#define WAVES_PER_BLOCK 8

typedef __attribute__((ext_vector_type(2))) float v2f;
typedef __attribute__((ext_vector_type(8))) float v8f;

union V8U { v8f v; float f[8]; };

// sentinel just below 25.0f: d2 > NEG_SENTINEL  <=>  d2 >= 25.0f (exact)
__device__ __forceinline__ float neg_sentinel() { return __uint_as_float(0x41C7FFFFu); }

// ---------------------------------------------------------------------------
// Kernel 1: WMMA-based mining. Each wave owns 16 anchor rows, scans one
// 2048-column split. dot(a_i, c_j) tiles come from V_WMMA_F32_16X16X4_F32
// (K=2 padded to 4); A is pre-scaled by -2 so d2 = |a|^2 + |c|^2 + acc.
// All candidate tracking is branchless (v_cndmask), no EXEC divergence.
// ---------------------------------------------------------------------------
__global__ __launch_bounds__(256)
void mine_partial_kernel(const float* __restrict__ coords,
                         const int*   __restrict__ aidx,
                         float* __restrict__ pvPart, int* __restrict__ pjPart,
                         float* __restrict__ nvPart, int* __restrict__ njPart)
{
    const int lane = threadIdx.x & 31;
    const int wv   = threadIdx.x >> 5;
    const int anchorBlk = blockIdx.x / JSPLIT;
    const int split     = blockIdx.x % JSPLIT;
    const int base = (anchorBlk * WAVES_PER_BLOCK + wv) * 16;
    const int hi = lane >> 4;     // 0: lanes 0-15, 1: lanes 16-31
    const int lo = lane & 15;

    const float2* c2 = (const float2*)coords;

    // anchor coordinates for row base+lo (all lanes use &15 index: no OOB)
    const int   self = aidx[base + lo];
    const float2 a   = c2[self];
    const float an2_own = a.x * a.x + a.y * a.y;

    // A operand: lanes 0-15 carry K=0,1 = -2*a ; lanes 16-31 = K=2,3 zero pad
    v2f A;
    A.x = hi ? 0.0f : (-2.0f * a.x);
    A.y = hi ? 0.0f : (-2.0f * a.y);

    // D-tile slot r belongs to row base + 8*hi + r  -> pull that row's metadata
    float an2r[8];
    int   selfr[8];
#pragma unroll
    for (int r = 0; r < 8; ++r) {
        an2r[r]  = __shfl(an2_own, 8 * hi + r, 32);
        selfr[r] = __shfl(self,    8 * hi + r, 32);
    }

    // pos: init 25.0 -> update rule (d2 < pv) also enforces d2 < 25
    // neg: init nextbelow(25) -> update rule (d2 > nv) enforces d2 >= 25
    float pv[8]; int pj[8]; float nv[8]; int nj[8];
#pragma unroll
    for (int r = 0; r < 8; ++r) {
        pv[r] = MINE_T2;        pj[r] = -1;
        nv[r] = neg_sentinel(); nj[r] = -1;
    }

    const v8f zacc = {};
    int j = split * JCHUNK + lo;          // this lane's column for every tile
    for (int t = 0; t < TILES_PER_SPLIT; ++t, j += 16) {
        const float2 c = c2[j];
        __builtin_prefetch(&c2[j + 16], 0, 3);
        const float cn2 = c.x * c.x + c.y * c.y;

        v2f B;                            // lanes 0-15: K=0,1 ; lanes 16-31: zero pad
        B.x = hi ? 0.0f : c.x;
        B.y = hi ? 0.0f : c.y;

        V8U acc; acc.v = zacc;
        acc.v = __builtin_amdgcn_wmma_f32_16x16x4_f32(
            /*neg_a=*/false, A, /*neg_b=*/false, B,
            /*c_mod=*/(short)0, acc.v, /*reuse_a=*/false, /*reuse_b=*/false);

#pragma unroll
        for (int r = 0; r < 8; ++r) {
            const float d2 = fmaxf(an2r[r] + cn2 + acc.f[r], 0.0f);
            // pos side: needs self exclusion (self distance == 0)
            const bool pUpd = (j != selfr[r]) & (d2 < pv[r]);
            pv[r] = pUpd ? d2 : pv[r];
            pj[r] = pUpd ? j  : pj[r];
            // neg side: self distance ~0 can never reach >= 25, no self check
            const bool nUpd = (d2 > nv[r]);
            nv[r] = nUpd ? d2 : nv[r];
            nj[r] = nUpd ? j  : nj[r];
        }
    }

    // reduce across the 16 lanes of each half; exact-tie -> smaller column index
    // (sentinel entries always carry index -1 and only tie with other sentinels)
#pragma unroll
    for (int r = 0; r < 8; ++r) {
#pragma unroll
        for (int m = 8; m >= 1; m >>= 1) {
            const float opv = __shfl_xor(pv[r], m, 32);
            const int   opj = __shfl_xor(pj[r], m, 32);
            const bool  pT  = (opv < pv[r]) | ((opv == pv[r]) & (opj < pj[r]) & (opj >= 0));
            pv[r] = pT ? opv : pv[r];
            pj[r] = pT ? opj : pj[r];
            const float onv = __shfl_xor(nv[r], m, 32);
            const int   onj = __shfl_xor(nj[r], m, 32);
            const bool  nT  = (onv > nv[r]) | ((onv == nv[r]) & (onj < nj[r]) & (onj >= 0));
            nv[r] = nT ? onv : nv[r];
            nj[r] = nT ? onj : nj[r];
        }
    }

    if (lo == 0) {
#pragma unroll
        for (int r = 0; r < 8; ++r) {
            const int row = base + 8 * hi + r;
            pvPart[row * JSPLIT + split] = pv[r];
            pjPart[row * JSPLIT + split] = pj[r];
            nvPart[row * JSPLIT + split] = nv[r];
            njPart[row * JSPLIT + split] = nj[r];
        }
    }
}

// ---------------------------------------------------------------------------
// Kernel 1b: merge the JSPLIT partial argmin/argmax (splits are ascending in
// j, tie -> smaller index == first occurrence). Branchless.
// ---------------------------------------------------------------------------
__global__ __launch_bounds__(256)
void mine_combine_kernel(const float* __restrict__ pvPart, const int* __restrict__ pjPart,
                         const float* __restrict__ nvPart, const int* __restrict__ njPart,
                         int* __restrict__ posIdx, int* __restrict__ negIdx)
{
    const int i = blockIdx.x * blockDim.x + threadIdx.x;
    float bpv = __builtin_inff(); int bpj = -1;
    float bnv = -1.0f;            int bnj = -1;
#pragma unroll
    for (int s = 0; s < JSPLIT; ++s) {
        const float v  = pvPart[i * JSPLIT + s];
        const int   jj = pjPart[i * JSPLIT + s];
        const bool  pT = (v < bpv) | ((v == bpv) & (jj < bpj) & (jj >= 0));
        bpv = pT ? v  : bpv;
        bpj = pT ? jj : bpj;
        const float w  = nvPart[i * JSPLIT + s];
        const int   kk = njPart[i * JSPLIT + s];
        const bool  nT = (w > bnv) | ((w == bnv) & (kk < bnj) & (kk >= 0));
        bnv = nT ? w  : bnv;
        bnj = nT ? kk : bnj;
    }
    posIdx[i] = bpj;
    negIdx[i] = bnj;
}

// ---------------------------------------------------------------------------
// Kernel 2: per-anchor triplet term. One wave per anchor; each lane owns 16
// of the 512 dims via float4 loads (fully coalesced 128B/clause per wave).
// ---------------------------------------------------------------------------
__global__ __launch_bounds__(256)
void triplet_kernel(const float* __restrict__ emb, const int* __restrict__ aidx,
                    const int* __restrict__ posIdx, const int* __restrict__ negIdx,
                    float* __restrict__ tl, float* __restrict__ validf)
{
    const int lane = threadIdx.x & 31;
    const int wv   = threadIdx.x >> 5;
    const int i    = blockIdx.x * WAVES_PER_BLOCK + wv;

    const int pi = posIdx[i];
    const int ni = negIdx[i];
    if (pi < 0 || ni < 0) {                 // wave-uniform branch
        if (lane == 0) { tl[i] = 0.0f; validf[i] = 0.0f; }
        return;
    }

    const float4* ea = (const float4*)(emb + (size_t)aidx[i] * DIMS);
    const float4* ep = (const float4*)(emb + (size_t)pi * DIMS);
    const float4* en = (const float4*)(emb + (size_t)ni * DIMS);

    float ps = 0.0f, ns = 0.0f;
#pragma unroll
    for (int c = 0; c < 4; ++c) {
        const int k = lane + 32 * c;
        const float4 fa = ea[k];
        const float4 fp = ep[k];
        const float4 fn = en[k];
        float d;
        d = fa.x - fp.x; ps += d * d;  d = fa.y - fp.y; ps += d * d;
        d = fa.z - fp.z; ps += d * d;  d = fa.w - fp.w; ps += d * d;
        d = fa.x - fn.x; ns += d * d;  d = fa.y - fn.y; ns += d * d;
        d = fa.z - fn.z; ns += d * d;  d = fa.w - fn.w; ns += d * d;
    }
#pragma unroll
    for (int m = 16; m >= 1; m >>= 1) {
        ps += __shfl_xor(ps, m, 32);
        ns += __shfl_xor(ns, m, 32);
    }
    if (lane == 0) {
        tl[i] = fmaxf(sqrtf(ps) - sqrtf(ns) + MARGIN, 0.0f);
        validf[i] = 1.0f;
    }
}

// ---------------------------------------------------------------------------
// Kernel 3: deterministic fixed-order final reduction.
// ---------------------------------------------------------------------------
__global__ __launch_bounds__(256)
void reduce_kernel(const float* __restrict__ tl, const float* __restrict__ validf,
                   float* __restrict__ out)
{
    __shared__ float ss[256];
    __shared__ float sc[256];
    float s = 0.0f, c = 0.0f;
    for (int k = threadIdx.x; k < NROWS; k += 256) { s += tl[k]; c += validf[k]; }
    ss[threadIdx.x] = s; sc[threadIdx.x] = c;
    __syncthreads();
    for (int st = 128; st > 0; st >>= 1) {
        if (threadIdx.x < (unsigned)st) {
            ss[threadIdx.x] += ss[threadIdx.x + st];
            sc[threadIdx.x] += sc[threadIdx.x + st];
        }
        __syncthreads();
    }
    if (threadIdx.x == 0) out[0] = ss[0] / fmaxf(sc[0], 1.0f);
}

// ---------------------------------------------------------------------------
extern "C" void kernel_launch(void* const* d_in, const int* in_sizes, int n_in,
                              void* d_out, int out_size, void* d_ws, size_t ws_size,
                              hipStream_t stream)
{
    const float* emb    = (const float*)d_in[0];   // [8192, 512] f32
    const float* coords = (const float*)d_in[1];   // [8192, 2]   f32
    const int*   aidx   = (const int*)  d_in[2];   // [8192]      i32
    float* out = (float*)d_out;

    char* ws = (char*)d_ws;
    int*   posIdx = (int*)(ws);
    int*   negIdx = (int*)(ws + 4 * NROWS);
    float* tl     = (float*)(ws + 8 * NROWS);
    float* validf = (float*)(ws + 12 * NROWS);
    float* pvPart = (float*)(ws + 16 * NROWS);
    int*   pjPart = (int*)(ws + 16 * NROWS + 4 * NROWS * JSPLIT);
    float* nvPart = (float*)(ws + 16 * NROWS + 8 * NROWS * JSPLIT);
    int*   njPart = (int*)(ws + 16 * NROWS + 12 * NROWS * JSPLIT);

    const int mineBlocks = (NROWS / (16 * WAVES_PER_BLOCK)) * JSPLIT;  // 256
    mine_partial_kernel<<<mineBlocks, 256, 0, stream>>>(coords, aidx,
                                                        pvPart, pjPart, nvPart, njPart);
    mine_combine_kernel<<<NROWS / 256, 256, 0, stream>>>(pvPart, pjPart, nvPart, njPart,
                                                         posIdx, negIdx);
    triplet_kernel<<<NROWS / WAVES_PER_BLOCK, 256, 0, stream>>>(emb, aidx, posIdx, negIdx,
                                                                tl, validf);
    reduce_kernel<<<1, 256, 0, stream>>>(tl, validf, out);
}